// NormSingleGAT_53549652247254
// MI455X (gfx1250) — compile-verified
//
#include <hip/hip_runtime.h>
#include <hip/hip_bf16.h>
#include <math.h>

// ---------------------------------------------------------------------------
// MI455X (gfx1250) implementation of RMSNorm + GATv2Conv.
//  - GEMMs use V_WMMA_F32_16X16X4_F32 (full fp32 matrix core path; workload is
//    memory-bound, so fp32 WMMA costs nothing vs f16 and matches reference).
//  - Edge phase: wave32-per-edge gathers, shfl reductions, f32 atomics.
// ---------------------------------------------------------------------------

#define N_DIM   128
#define HC_DIM  256   // H*C = 2*128
#define NEG_SLOPE 0.2f
#define RMS_EPS   1e-6f

typedef float v2f __attribute__((ext_vector_type(2)));
typedef float v8f __attribute__((ext_vector_type(8)));

__device__ __forceinline__ float waveReduceSum(float v) {
    #pragma unroll
    for (int m = 16; m >= 1; m >>= 1) v += __shfl_xor(v, m, 32);
    return v;
}

// Monotonic float atomic-max via integer atomics (sign-aware IEEE ordering).
__device__ __forceinline__ void atomicMaxF32(float* addr, float val) {
    if (val >= 0.0f) {
        atomicMax((int*)addr, __float_as_int(val));
    } else {
        atomicMin((unsigned int*)addr, __float_as_uint(val));
    }
}

// ---------------------------------------------------------------------------
// Init: zero output + denom, set amax = -inf-ish
// ---------------------------------------------------------------------------
__global__ void init_kernel(float* __restrict__ out, float* __restrict__ amax,
                            float* __restrict__ denom, int nOut, int nNH) {
    int t = blockIdx.x * blockDim.x + threadIdx.x;
    if (t < nOut) out[t] = 0.0f;
    if (t < nNH) { amax[t] = -3.0e38f; denom[t] = 0.0f; }
}

// ---------------------------------------------------------------------------
// RMSNorm: one wave per row (D=128 -> float4 per lane)
// ---------------------------------------------------------------------------
__global__ void rmsnorm_kernel(const float* __restrict__ x,
                               const float* __restrict__ w,
                               float* __restrict__ xn, int N) {
    int wid  = (blockIdx.x * blockDim.x + threadIdx.x) >> 5;
    int lane = threadIdx.x & 31;
    if (wid >= N) return;
    const float4* xr = (const float4*)(x + (size_t)wid * N_DIM);
    const float4* wr = (const float4*)w;
    float4 v = xr[lane];
    float ss = v.x*v.x + v.y*v.y + v.z*v.z + v.w*v.w;
    ss = waveReduceSum(ss);
    float r = rsqrtf(ss * (1.0f / N_DIM) + RMS_EPS);
    float4 g = wr[lane];
    float4 o;
    o.x = v.x * r * g.x; o.y = v.y * r * g.y;
    o.z = v.z * r * g.z; o.w = v.w * r * g.w;
    ((float4*)(xn + (size_t)wid * N_DIM))[lane] = o;
}

// ---------------------------------------------------------------------------
// fp32 WMMA GEMM: C[M,256] = A[M,128] * B[128,256]
// One wave computes a 16x64 strip: 4 adjacent 16x16 tiles (A regs reused x4).
// K swept in steps of 4 via V_WMMA_F32_16X16X4_F32.
//   A 16x4 layout:  lanes 0-15 K=0/1 in v0/v1, lanes 16-31 K=2/3.
//   B 4x16 layout:  lane = N; v0/v1 = K (0/1 low half, 2/3 high half).
//   C/D: VGPR v -> rows v (lanes 0-15) and v+8 (lanes 16-31), N = lane&15.
// ---------------------------------------------------------------------------
#define GEMM_NT 4   // n-tiles per wave (64 columns)

__global__ void gemm_wmma_f32_kernel(const float* __restrict__ A,
                                     const float* __restrict__ B,
                                     float* __restrict__ C, int M) {
    const int wid  = (blockIdx.x * blockDim.x + threadIdx.x) >> 5;
    const int lane = threadIdx.x & 31;
    const int wavesPerRow = HC_DIM / (16 * GEMM_NT);        // 4
    const int tileM = wid / wavesPerRow;
    const int nt0   = (wid % wavesPerRow) * GEMM_NT;        // first 16-wide tile
    if (tileM * 16 >= M) return;

    const int rlane = lane & 15;
    const int khi   = (lane >> 4) << 1;                     // 0 or 2
    const float* arow = A + (size_t)(tileM * 16 + rlane) * N_DIM + khi;

    v8f acc[GEMM_NT];
    #pragma unroll
    for (int t = 0; t < GEMM_NT; ++t) acc[t] = (v8f){};

    #pragma unroll
    for (int k = 0; k < N_DIM; k += 4) {
        v2f a;
        a.x = arow[k];
        a.y = arow[k + 1];
        const float* brow = B + (size_t)(k + khi) * HC_DIM;
        #pragma unroll
        for (int t = 0; t < GEMM_NT; ++t) {
            const int col = (nt0 + t) * 16 + rlane;
            v2f b;
            b.x = brow[col];
            b.y = brow[col + HC_DIM];
            acc[t] = __builtin_amdgcn_wmma_f32_16x16x4_f32(
                /*neg_a=*/false, a, /*neg_b=*/false, b,
                /*c_mod=*/(short)0, acc[t],
                /*reuse_a=*/false, /*reuse_b=*/false);
        }
    }

    const int mrow = tileM * 16 + ((lane >> 4) << 3);       // +8 for high half
    #pragma unroll
    for (int t = 0; t < GEMM_NT; ++t) {
        const int col = (nt0 + t) * 16 + rlane;
        #pragma unroll
        for (int v = 0; v < 8; ++v) {
            C[(size_t)(mrow + v) * HC_DIM + col] = acc[t][v];
        }
    }
}

// ---------------------------------------------------------------------------
// Edge pass 1: alpha[e,h] = att_h . leaky(x_l[src] + x_r[dst]); segment max
// One wave per edge; lane covers 4 channels per head.
// ---------------------------------------------------------------------------
__global__ void edge_alpha_kernel(const float* __restrict__ xl,
                                  const float* __restrict__ xr,
                                  const int* __restrict__ ei,   // [2,E]
                                  const float* __restrict__ att,
                                  float* __restrict__ alpha,    // [E,2]
                                  float* __restrict__ amax,     // [N,2]
                                  int E) {
    int wid  = (blockIdx.x * blockDim.x + threadIdx.x) >> 5;
    int lane = threadIdx.x & 31;
    if (wid >= E) return;
    int s = ei[wid], d = ei[E + wid];
    const float4* xls = (const float4*)(xl + (size_t)s * HC_DIM);
    const float4* xrd = (const float4*)(xr + (size_t)d * HC_DIM);
    const float4* at4 = (const float4*)att;

    float sums[2];
    #pragma unroll
    for (int h = 0; h < 2; ++h) {
        float4 a = xls[h * 32 + lane];
        float4 b = xrd[h * 32 + lane];
        float4 w = at4[h * 32 + lane];
        float m0 = a.x + b.x; m0 = (m0 > 0.0f) ? m0 : NEG_SLOPE * m0;
        float m1 = a.y + b.y; m1 = (m1 > 0.0f) ? m1 : NEG_SLOPE * m1;
        float m2 = a.z + b.z; m2 = (m2 > 0.0f) ? m2 : NEG_SLOPE * m2;
        float m3 = a.w + b.w; m3 = (m3 > 0.0f) ? m3 : NEG_SLOPE * m3;
        float sl = m0 * w.x + m1 * w.y + m2 * w.z + m3 * w.w;
        sums[h] = waveReduceSum(sl);
    }
    if (lane == 0) {
        alpha[(size_t)wid * 2]     = sums[0];
        alpha[(size_t)wid * 2 + 1] = sums[1];
        atomicMaxF32(&amax[(size_t)d * 2],     sums[0]);
        atomicMaxF32(&amax[(size_t)d * 2 + 1], sums[1]);
    }
}

// ---------------------------------------------------------------------------
// Edge pass 2: ex = exp(alpha - amax[dst]); denom[dst] += ex (in-place alpha)
// One thread per (edge, head).
// ---------------------------------------------------------------------------
__global__ void edge_exp_kernel(float* __restrict__ alpha,       // [E,2] in/out
                                const int* __restrict__ ei,
                                const float* __restrict__ amax,  // [N,2]
                                float* __restrict__ denom,       // [N,2]
                                int E) {
    int t = blockIdx.x * blockDim.x + threadIdx.x;
    if (t >= 2 * E) return;
    int e = t >> 1, h = t & 1;
    int d = ei[E + e];
    float ex = __expf(alpha[t] - amax[(size_t)d * 2 + h]);
    alpha[t] = ex;
    atomicAdd(&denom[(size_t)d * 2 + h], ex);
}

// ---------------------------------------------------------------------------
// Edge pass 3: out[dst] += 0.5 * (w0 * xl[src,h0] + w1 * xl[src,h1])
// One wave per edge; head-mean folded before the atomic (128 atomics/edge).
// ---------------------------------------------------------------------------
__global__ void edge_scatter_kernel(const float* __restrict__ xl,
                                    const int* __restrict__ ei,
                                    const float* __restrict__ exw,   // [E,2]
                                    const float* __restrict__ denom, // [N,2]
                                    float* __restrict__ out,         // [N,128]
                                    int E) {
    int wid  = (blockIdx.x * blockDim.x + threadIdx.x) >> 5;
    int lane = threadIdx.x & 31;
    if (wid >= E) return;
    int s = ei[wid], d = ei[E + wid];
    float w0 = exw[(size_t)wid * 2]     / (denom[(size_t)d * 2]     + 1e-16f);
    float w1 = exw[(size_t)wid * 2 + 1] / (denom[(size_t)d * 2 + 1] + 1e-16f);
    const float4* xls = (const float4*)(xl + (size_t)s * HC_DIM);
    float4 a0 = xls[lane];        // head 0, channels lane*4..+3
    float4 a1 = xls[32 + lane];   // head 1
    float* o = out + (size_t)d * 128 + lane * 4;
    atomicAdd(o + 0, 0.5f * (w0 * a0.x + w1 * a1.x));
    atomicAdd(o + 1, 0.5f * (w0 * a0.y + w1 * a1.y));
    atomicAdd(o + 2, 0.5f * (w0 * a0.z + w1 * a1.z));
    atomicAdd(o + 3, 0.5f * (w0 * a0.w + w1 * a1.w));
}

// ---------------------------------------------------------------------------
// Launch
// ---------------------------------------------------------------------------
extern "C" void kernel_launch(void* const* d_in, const int* in_sizes, int n_in,
                              void* d_out, int out_size, void* d_ws, size_t ws_size,
                              hipStream_t stream) {
    const float* x     = (const float*)d_in[0];   // [N,128]
    const int*   ei    = (const int*)  d_in[1];   // [2,E]
    const float* rmsw  = (const float*)d_in[2];   // [128]
    const float* W_l   = (const float*)d_in[3];   // [128,256]
    const float* W_r   = (const float*)d_in[4];   // [128,256]
    const float* att   = (const float*)d_in[5];   // [2,128]

    const int N = in_sizes[0] / N_DIM;            // 50000
    const int E = in_sizes[1] / 2;                // 600000

    // Workspace layout (floats)
    float* ws    = (float*)d_ws;
    float* xn    = ws;                                  // N*128
    float* xl    = xn + (size_t)N * N_DIM;              // N*256
    float* xr    = xl + (size_t)N * HC_DIM;             // N*256
    float* alpha = xr + (size_t)N * HC_DIM;             // E*2
    float* amax  = alpha + (size_t)E * 2;               // N*2
    float* denom = amax  + (size_t)N * 2;               // N*2

    float* out = (float*)d_out;                         // [N,128]

    // init
    {
        int total = N * 128;  // >= N*2
        init_kernel<<<(total + 255) / 256, 256, 0, stream>>>(out, amax, denom,
                                                             total, N * 2);
    }
    // rmsnorm: 8 waves/block -> 8 rows/block
    rmsnorm_kernel<<<(N + 7) / 8, 256, 0, stream>>>(x, rmsw, xn, N);

    // GEMMs: one wave per 16x64 strip
    {
        int totalWaves = ((N + 15) / 16) * (HC_DIM / (16 * GEMM_NT));
        int blocks = (totalWaves + 7) / 8;
        gemm_wmma_f32_kernel<<<blocks, 256, 0, stream>>>(xn, W_l, xl, N);
        gemm_wmma_f32_kernel<<<blocks, 256, 0, stream>>>(xn, W_r, xr, N);
    }

    // edge phase
    edge_alpha_kernel<<<(E + 7) / 8, 256, 0, stream>>>(xl, xr, ei, att,
                                                       alpha, amax, E);
    edge_exp_kernel<<<(2 * E + 255) / 256, 256, 0, stream>>>(alpha, ei, amax,
                                                             denom, E);
    edge_scatter_kernel<<<(E + 7) / 8, 256, 0, stream>>>(xl, ei, alpha, denom,
                                                         out, E);
}